// GCN_15496242004677
// MI455X (gfx1250) — compile-verified
//
#include <hip/hip_runtime.h>
#include <hip/hip_bf16.h>

typedef float v2f __attribute__((ext_vector_type(2)));
typedef float v8f __attribute__((ext_vector_type(8)));

#define D_FEAT 64
#define D_OUT  32

// ---------------------------------------------------------------- utilities
__global__ void fill_f32(float* __restrict__ p, float v, int n) {
    int i = blockIdx.x * blockDim.x + threadIdx.x;
    if (i < n) p[i] = v;
}

__global__ void deg_count(const int* __restrict__ dst, float* __restrict__ deg, int e) {
    int i = blockIdx.x * blockDim.x + threadIdx.x;
    if (i < e) unsafeAtomicAdd(&deg[dst[i]], 1.0f);
}

__global__ void rsqrt_inplace(float* __restrict__ p, int n) {
    int i = blockIdx.x * blockDim.x + threadIdx.x;
    if (i < n) {
        float d = p[i];
        p[i] = d > 0.0f ? __frsqrt_rn(d) : 0.0f;
    }
}

// ------------------------------------------------------- GEMM: H = X * W (N x 64 @ 64 x 64)
// One wave per 16-row strip; 4 f32 accumulators cover all 64 output columns.
__global__ __launch_bounds__(256) void gcn_gemm64_wmma(
    const float* __restrict__ X, const float* __restrict__ W,
    float* __restrict__ H, int nrows)
{
    const int lane  = threadIdx.x & 31;
    const int strip = blockIdx.x * 8 + (threadIdx.x >> 5);
    const int r0 = strip * 16;
    if (r0 >= nrows) return;

    const int m  = lane & 15;            // A: row within tile
    const int kk = (lane >> 4) * 2;      // A/B: k-offset within 4-wide k-step
    const int n  = lane & 15;            // B: col within 16-wide tile

    v8f acc0 = {}, acc1 = {}, acc2 = {}, acc3 = {};
    const float* xrow = X + (size_t)(r0 + m) * D_FEAT;

    #pragma unroll
    for (int k0 = 0; k0 < D_FEAT; k0 += 4) {
        v2f a;
        a.x = xrow[k0 + kk];
        a.y = xrow[k0 + kk + 1];
        const float* w0 = W + (size_t)(k0 + kk) * D_FEAT + n;
        const float* w1 = W + (size_t)(k0 + kk + 1) * D_FEAT + n;
        v2f b0; b0.x = w0[0];  b0.y = w1[0];
        v2f b1; b1.x = w0[16]; b1.y = w1[16];
        v2f b2; b2.x = w0[32]; b2.y = w1[32];
        v2f b3; b3.x = w0[48]; b3.y = w1[48];
        acc0 = __builtin_amdgcn_wmma_f32_16x16x4_f32(false, a, false, b0, (short)0, acc0, false, false);
        acc1 = __builtin_amdgcn_wmma_f32_16x16x4_f32(false, a, false, b1, (short)0, acc1, false, false);
        acc2 = __builtin_amdgcn_wmma_f32_16x16x4_f32(false, a, false, b2, (short)0, acc2, false, false);
        acc3 = __builtin_amdgcn_wmma_f32_16x16x4_f32(false, a, false, b3, (short)0, acc3, false, false);
    }

    const int rbase = (lane >> 4) * 8;   // C/D: lanes 16-31 hold rows 8..15
    const int col   = lane & 15;
    #pragma unroll
    for (int r = 0; r < 8; ++r) {
        float* hrow = H + (size_t)(r0 + rbase + r) * D_FEAT + col;
        hrow[0]  = acc0[r];
        hrow[16] = acc1[r];
        hrow[32] = acc2[r];
        hrow[48] = acc3[r];
    }
}

// ------------------------------------------------------- edge scatter: agg[dst] += norm * h[src]
// 16 threads per edge, float4 per thread (64 features).
__global__ __launch_bounds__(256) void gcn_scatter(
    const int* __restrict__ srcIdx, const int* __restrict__ dstIdx,
    const float* __restrict__ dinv, const float* __restrict__ h,
    float* __restrict__ agg, int nEdges)
{
    int idx = blockIdx.x * blockDim.x + threadIdx.x;
    if (idx >= nEdges * 16) return;
    int e = idx >> 4;
    int f = (idx & 15) * 4;
    int s = srcIdx[e], d = dstIdx[e];
    float nrm = dinv[s] * dinv[d];
    const float4 hv = *reinterpret_cast<const float4*>(h + (size_t)s * D_FEAT + f);
    float* ap = agg + (size_t)d * D_FEAT + f;
    unsafeAtomicAdd(ap + 0, nrm * hv.x);
    unsafeAtomicAdd(ap + 1, nrm * hv.y);
    unsafeAtomicAdd(ap + 2, nrm * hv.z);
    unsafeAtomicAdd(ap + 3, nrm * hv.w);
}

// ------------------------------------------------------- combine: self-loop + bias + relu + residual
__global__ __launch_bounds__(256) void gcn_combine(
    const float* __restrict__ agg, const float* __restrict__ h,
    const float* __restrict__ dinv, const float* __restrict__ bias,
    const float* __restrict__ xprev, float* __restrict__ xnew, int n64)
{
    int idx = blockIdx.x * blockDim.x + threadIdx.x;
    if (idx >= n64) return;
    int node = idx >> 6, f = idx & 63;
    float di = dinv[node];
    float v = agg[idx] + di * di * h[idx] + bias[f];
    v = v > 0.0f ? v : 0.0f;
    xnew[idx] = v + xprev[idx];
}

// ------------------------------------------------------- pooling
__global__ void pool_count(const int* __restrict__ batch, float* __restrict__ cnts, int n) {
    int i = blockIdx.x * blockDim.x + threadIdx.x;
    if (i < n) unsafeAtomicAdd(&cnts[batch[i]], 1.0f);
}

__global__ __launch_bounds__(256) void pool_sum(
    const float* __restrict__ x, const int* __restrict__ batch,
    float* __restrict__ sums, int n64)
{
    int idx = blockIdx.x * blockDim.x + threadIdx.x;
    if (idx >= n64) return;
    int node = idx >> 6, f = idx & 63;
    unsafeAtomicAdd(&sums[(size_t)batch[node] * D_FEAT + f], x[idx]);
}

// ------------------------------------------------------- final: out = (sums/cnt) @ Wl + bl  (G x 64 @ 64 x 32)
__global__ __launch_bounds__(256) void final_linear_wmma(
    const float* __restrict__ sums, const float* __restrict__ cnts,
    const float* __restrict__ Wl, const float* __restrict__ bl,
    float* __restrict__ out, int nrows)
{
    const int lane  = threadIdx.x & 31;
    const int strip = blockIdx.x * 8 + (threadIdx.x >> 5);
    const int r0 = strip * 16;
    if (r0 >= nrows) return;

    const int m  = lane & 15;
    const int kk = (lane >> 4) * 2;
    const int n  = lane & 15;
    const int row = r0 + m;
    float c = cnts[row];
    float inv = 1.0f / (c > 1.0f ? c : 1.0f);

    v8f acc0 = {}, acc1 = {};
    const float* srow = sums + (size_t)row * D_FEAT;

    #pragma unroll
    for (int k0 = 0; k0 < D_FEAT; k0 += 4) {
        v2f a;
        a.x = srow[k0 + kk] * inv;
        a.y = srow[k0 + kk + 1] * inv;
        const float* w0 = Wl + (size_t)(k0 + kk) * D_OUT + n;
        const float* w1 = Wl + (size_t)(k0 + kk + 1) * D_OUT + n;
        v2f b0; b0.x = w0[0];  b0.y = w1[0];
        v2f b1; b1.x = w0[16]; b1.y = w1[16];
        acc0 = __builtin_amdgcn_wmma_f32_16x16x4_f32(false, a, false, b0, (short)0, acc0, false, false);
        acc1 = __builtin_amdgcn_wmma_f32_16x16x4_f32(false, a, false, b1, (short)0, acc1, false, false);
    }

    const int rbase = (lane >> 4) * 8;
    const int col   = lane & 15;
    #pragma unroll
    for (int r = 0; r < 8; ++r) {
        float* orow = out + (size_t)(r0 + rbase + r) * D_OUT + col;
        orow[0]  = acc0[r] + bl[col];
        orow[16] = acc1[r] + bl[col + 16];
    }
}

// ---------------------------------------------------------------- host launcher
static inline int cdiv(long long a, int b) { return (int)((a + b - 1) / b); }

extern "C" void kernel_launch(void* const* d_in, const int* in_sizes, int n_in,
                              void* d_out, int out_size, void* d_ws, size_t ws_size,
                              hipStream_t stream)
{
    const float* x     = (const float*)d_in[0];
    const int*   ei    = (const int*)d_in[1];
    const int*   batch = (const int*)d_in[2];
    const float* W[3]  = {(const float*)d_in[3], (const float*)d_in[5], (const float*)d_in[7]};
    const float* B[3]  = {(const float*)d_in[4], (const float*)d_in[6], (const float*)d_in[8]};
    const float* Wl    = (const float*)d_in[9];
    const float* bl    = (const float*)d_in[10];
    float* out = (float*)d_out;

    const int N = in_sizes[0] / D_FEAT;   // 100000 (divisible by 16)
    const int E = in_sizes[1] / 2;        // 1600000
    const int G = out_size / D_OUT;       // 256
    const int* srcIdx = ei;
    const int* dstIdx = ei + E;

    float* ws   = (float*)d_ws;
    float* buf0 = ws;                          // N*64
    float* buf1 = buf0 + (size_t)N * D_FEAT;   // N*64
    float* hbuf = buf1 + (size_t)N * D_FEAT;   // N*64
    float* dinv = hbuf + (size_t)N * D_FEAT;   // N   (deg then dinv, in place)
    float* sums = dinv + N;                    // G*64
    float* cnts = sums + (size_t)G * D_FEAT;   // G

    const int TB = 256;
    const int n64 = N * D_FEAT;

    // degree (self-loop counts as 1) -> dinv
    fill_f32<<<cdiv(N, TB), TB, 0, stream>>>(dinv, 1.0f, N);
    deg_count<<<cdiv(E, TB), TB, 0, stream>>>(dstIdx, dinv, E);
    rsqrt_inplace<<<cdiv(N, TB), TB, 0, stream>>>(dinv, N);

    // zero pooling buffers
    fill_f32<<<cdiv(G * D_FEAT, TB), TB, 0, stream>>>(sums, 0.0f, G * D_FEAT);
    fill_f32<<<cdiv(G, TB), TB, 0, stream>>>(cnts, 0.0f, G);

    const int gemmBlocks = cdiv((long long)cdiv(N, 16), 8);
    const float* xcur = x;
    float* aggbuf[3] = {buf0, buf1, buf0};

    for (int l = 0; l < 3; ++l) {
        float* agg = aggbuf[l];
        gcn_gemm64_wmma<<<gemmBlocks, TB, 0, stream>>>(xcur, W[l], hbuf, N);
        fill_f32<<<cdiv(n64, TB), TB, 0, stream>>>(agg, 0.0f, n64);
        gcn_scatter<<<cdiv((long long)E * 16, TB), TB, 0, stream>>>(srcIdx, dstIdx, dinv, hbuf, agg, E);
        gcn_combine<<<cdiv(n64, TB), TB, 0, stream>>>(agg, hbuf, dinv, B[l], xcur, agg, n64);
        xcur = agg;
    }

    // global mean pool + final linear
    pool_count<<<cdiv(N, TB), TB, 0, stream>>>(batch, cnts, N);
    pool_sum<<<cdiv(n64, TB), TB, 0, stream>>>(xcur, batch, sums, n64);
    final_linear_wmma<<<cdiv((long long)cdiv(G, 16), 8), TB, 0, stream>>>(sums, cnts, Wl, bl, out, G);
}